// RationalQuadraticSplineAutoregressiveSubsetTransform2d_69217692942840
// MI455X (gfx1250) — compile-verified
//
#include <hip/hip_runtime.h>
#include <hip/hip_bf16.h>

// Fused RQ-spline transform for MI455X (gfx1250, wave32).
//
// Shapes (fixed by the reference): B=16, C=3, H=W=256, K=8, P=25, O=C*P=75.
// Memory-bound: ~50MB total traffic -> ~2.2us floor at 23.3 TB/s. The fusion
// keeps the (B,75,H,W) params tensor in LDS instead of HBM (saves ~630MB of
// traffic vs the reference dataflow). The per-pixel 75x3 GEMM runs on
// V_WMMA_F32_16X16X4_F32 (fp32 accuracy required by softmax/cumsum): 5 row
// tiles x 2 column halves = 10 WMMAs per wave per 32 spatial positions.
// All operand builds are branch-free so EXEC is all-ones across the WMMA loop.

typedef __attribute__((ext_vector_type(2))) float v2f;
typedef __attribute__((ext_vector_type(8))) float v8f;

#define BLOCK   128
#define HW      65536      // H*W
#define CH      3
#define KB      8          // spline bins
#define PP      25         // params per channel (3K+1)
#define OROWS   75         // C*PP
#define OPAD    80         // padded to 5 WMMA row tiles
#define PSTRIDE 81         // LDS row stride in floats; 81 mod 64 = 17, gcd(17,64)=1

__device__ __forceinline__ float softplus_stable(float u) {
    // log1p(exp(u)) = max(u,0) + log(1 + exp(-|u|))
    return fmaxf(u, 0.0f) + __logf(1.0f + __expf(-fabsf(u)));
}

__launch_bounds__(BLOCK)
__global__ void rq_spline_fused(const float* __restrict__ xl,
                                const float* __restrict__ xu,
                                const float* __restrict__ Wm,
                                const float* __restrict__ bv,
                                float* __restrict__ out,
                                int ntot) {
    __shared__ float Wlds[OPAD * 4];          // W padded: [o][k], k=3 pad and o>=75 are 0
    __shared__ float xstage[BLOCK][4];        // per-position x[3] (+pad 0 for K=3)
    __shared__ float pLds[BLOCK * PSTRIDE];   // per-position 80 params (75 used)

    const int  tid   = threadIdx.x;
    const long s     = (long)blockIdx.x * BLOCK + tid;  // spatial index in [0, B*HW)
    const long b_idx = s / HW;
    const long hw    = s - b_idx * HW;
    const long base  = b_idx * (long)(CH * HW) + hw;    // &x[b, 0, h, w]

    // ---- stage W into LDS, padded to [80][4] with zeros (branch-free A builds) ----
    for (int i = tid; i < OPAD * 4; i += BLOCK) {
        const int o = i >> 2;
        const int k = i & 3;
        Wlds[i] = (o < OROWS && k < CH) ? Wm[o * CH + k] : 0.0f;
    }

    // ---- stage this position's x_lower channels ----
    const float xl0 = xl[base];
    const float xl1 = xl[base + HW];
    const float xl2 = xl[base + 2 * HW];
    xstage[tid][0] = xl0;
    xstage[tid][1] = xl1;
    xstage[tid][2] = xl2;
    xstage[tid][3] = 0.0f;                     // K=3 pad column
    __syncthreads();

    // ---- WMMA phase: params[o, pos] = sum_k W[o,k] * x[k, pos] ----
    // A (16x4 fp32, documented layout): VGPR0 = K0 (lanes 0-15) / K2 (16-31),
    //                                   VGPR1 = K1 (lanes 0-15) / K3 (16-31)
    //   -> lane reads the contiguous pair Wlds[o*4 + (hi?2:0)] as one b64.
    // B (4x16 fp32): rows striped across lane halves, VGPR0 = K0/K1, VGPR1 = K2/K3.
    const int  wave = tid >> 5;
    const int  lane = tid & 31;
    const int  n    = lane & 15;
    const bool hi   = (lane >= 16);
    const int  aoff = hi ? 2 : 0;

    #pragma unroll
    for (int half = 0; half < 2; ++half) {
        const int colPos = (wave << 5) + (half << 4) + n;  // position this column maps to
        v2f Bm;
        Bm.x = xstage[colPos][hi ? 1 : 0];
        Bm.y = xstage[colPos][hi ? 3 : 2];
        #pragma unroll
        for (int t = 0; t < 5; ++t) {          // 5 row tiles cover 80 (>=75) outputs
            const int o = t * 16 + n;
            const float* Wq = &Wlds[(o << 2) + aoff];      // 8B-aligned pair
            v2f Am;
            Am.x = Wq[0];
            Am.y = Wq[1];
            v8f Cm = {};
            v8f D = __builtin_amdgcn_wmma_f32_16x16x4_f32(
                        false, Am, false, Bm, (short)0, Cm, false, false);
            // D layout: lane<16 holds rows M=r, lane>=16 rows M=8+r, col N=lane&15
            const int mBase = t * 16 + (hi ? 8 : 0);
            float* dst = &pLds[colPos * PSTRIDE + mBase];
            #pragma unroll
            for (int r = 0; r < 8; ++r) dst[r] = D[r];     // rows 75..79 = pad scratch
        }
    }
    __syncthreads();

    // ---- spline phase: each thread evaluates its own position, 3 channels ----
    const float* myP = &pLds[tid * PSTRIDE];
    const float  xls[3] = { xl0, xl1, xl2 };

    #pragma unroll
    for (int c = 0; c < CH; ++c) {
        const float* pp = myP + c * PP;
        const float* bb = bv  + c * PP;

        // softmax for widths and heights (stable)
        float uw[KB], uh[KB];
        float mw = -1e30f, mh = -1e30f;
        #pragma unroll
        for (int j = 0; j < KB; ++j) {
            uw[j] = pp[j] + bb[j];
            uh[j] = pp[KB + j] + bb[KB + j];
            mw = fmaxf(mw, uw[j]);
            mh = fmaxf(mh, uh[j]);
        }
        float sw = 0.0f, sh = 0.0f;
        #pragma unroll
        for (int j = 0; j < KB; ++j) {
            uw[j] = __expf(uw[j] - mw); sw += uw[j];
            uh[j] = __expf(uh[j] - mh); sh += uh[j];
        }
        const float rw = 1.0f / sw, rh = 1.0f / sh;

        // cumwidths / cumheights with pinned endpoints
        float cw[KB + 1], chh[KB + 1];
        cw[0] = 0.0f; chh[0] = 0.0f;
        float aw = 0.0f, ah = 0.0f;
        #pragma unroll
        for (int j = 0; j < KB; ++j) {
            aw += 0.001f + 0.992f * (uw[j] * rw);   // MIN_BIN_WIDTH + (1-K*MBW)*softmax
            ah += 0.001f + 0.992f * (uh[j] * rh);
            cw[j + 1]  = aw;
            chh[j + 1] = ah;
        }
        cw[KB] = 1.0f; chh[KB] = 1.0f;

        // derivatives = MIN_DERIVATIVE + softplus(ud)
        float dv[KB + 1];
        #pragma unroll
        for (int j = 0; j <= KB; ++j)
            dv[j] = 0.001f + softplus_stable(pp[2 * KB + j] + bb[2 * KB + j]);

        const float xin[2] = { xls[c], xu[base + (long)c * HW] };
        #pragma unroll
        for (int q = 0; q < 2; ++q) {
            const float xc = fminf(fmaxf(xin[q], 0.0f), 1.0f);

            // bin select via unrolled cndmask chain (register-resident, no scratch)
            float cwl = cw[0],  cwr = cw[1];
            float chl = chh[0], chr = chh[1];
            float d0  = dv[0],  d1  = dv[1];
            #pragma unroll
            for (int i = 1; i < KB; ++i) {
                const bool take = (xc >= cw[i]);
                cwl = take ? cw[i]      : cwl;
                cwr = take ? cw[i + 1]  : cwr;
                chl = take ? chh[i]     : chl;
                chr = take ? chh[i + 1] : chr;
                d0  = take ? dv[i]      : d0;
                d1  = take ? dv[i + 1]  : d1;
            }

            const float wbin  = cwr - cwl;
            const float hbin  = chr - chl;
            const float delta = hbin / wbin;
            const float th    = (xc - cwl) / wbin;
            const float t1m   = th * (1.0f - th);
            const float num   = hbin * (delta * th * th + d0 * t1m);
            const float den   = delta + (d0 + d1 - 2.0f * delta) * t1m;
            const float z     = chl + num / den;

            out[(q ? (long)ntot : 0L) + base + (long)c * HW] = z;
        }
    }
}

extern "C" void kernel_launch(void* const* d_in, const int* in_sizes, int n_in,
                              void* d_out, int out_size, void* d_ws, size_t ws_size,
                              hipStream_t stream) {
    const float* xl = (const float*)d_in[0];   // x_lower (B,C,H,W) fp32
    const float* xu = (const float*)d_in[1];   // x_upper (B,C,H,W) fp32
    const float* Wm = (const float*)d_in[2];   // W (75,3) fp32
    const float* bv = (const float*)d_in[3];   // b (75,)  fp32

    float* out = (float*)d_out;                // [z_lower | z_upper], each ntot floats
    const int ntot    = in_sizes[0];           // B*C*H*W
    const int spatial = ntot / CH;             // B*H*W
    const int grid    = (spatial + BLOCK - 1) / BLOCK;

    rq_spline_fused<<<grid, BLOCK, 0, stream>>>(xl, xu, Wm, bv, out, ntot);
}